// MemoryManager_50946902065314
// MI455X (gfx1250) — compile-verified
//
#include <hip/hip_runtime.h>

typedef __attribute__((ext_vector_type(2))) float v2f;
typedef __attribute__((ext_vector_type(8))) float v8f;

#define HEADS 32
#define DIM   128        // C / HEADS
#define CCH   4096       // channels
#define NB    8          // batch

// --------------------------------------------------------------------------
// metric[tok][d] = normalize_128( sum_h x[tok][h*128+d] )
// (mean over heads and the xw/size cancel under L2 normalization)
// --------------------------------------------------------------------------
__global__ void tome_metric(const float* __restrict__ x, float* __restrict__ metric) {
  const int tok = blockIdx.x;          // flat b*P + p
  const int d   = threadIdx.x;         // 0..127
  const float* row = x + (size_t)tok * CCH;
  float s = 0.0f;
#pragma unroll
  for (int h = 0; h < HEADS; ++h) s += row[h * DIM + d];

  __shared__ float red[DIM];
  red[d] = s * s;
  __syncthreads();
#pragma unroll
  for (int off = DIM / 2; off > 0; off >>= 1) {
    if (d < off) red[d] += red[d + off];
    __syncthreads();
  }
  const float rn = rsqrtf(red[0]);
  metric[(size_t)tok * DIM + d] = s * rn;
}

// --------------------------------------------------------------------------
// scores = am . bm^T, fused row-argmax.  One wave per 16 rows of am.
// FP32 WMMA 16x16x4, K = 128 => 32 v_wmma per 16x16 tile.
// --------------------------------------------------------------------------
__global__ void tome_scores_argmax(const float* __restrict__ metric,
                                   int* __restrict__ node_idx, int P) {
  const int pHalf = P >> 1;
  const int bb    = blockIdx.y;
  const int i0    = blockIdx.x * 16;       // am row block
  const int lane  = threadIdx.x;           // 0..31 (one wave)
  const int m     = lane & 15;
  const int hi    = lane >> 4;

  __shared__ __align__(16) float Ald[16][132];
  __shared__ __align__(16) float Bld[16][132];
  __shared__ float sc[16][17];

  const float* mb = metric + (size_t)bb * P * DIM;

  // stage A tile: rows are even tokens 2*(i0+r), 16 x 128 fp32
#pragma unroll
  for (int t = lane; t < 16 * 32; t += 32) {
    const int r  = t >> 5;
    const int c4 = (t & 31) << 2;
    const float4 v = *(const float4*)(mb + (size_t)(2 * (i0 + r)) * DIM + c4);
    *(float4*)(&Ald[r][c4]) = v;
  }

  float bestVal = -3.0e38f;
  int   bestIdx = 0;

  for (int j0 = 0; j0 < pHalf; j0 += 16) {
    // stage B tile: odd tokens 2*(j0+r)+1
#pragma unroll
    for (int t = lane; t < 16 * 32; t += 32) {
      const int r  = t >> 5;
      const int c4 = (t & 31) << 2;
      const float4 v = *(const float4*)(mb + (size_t)(2 * (j0 + r) + 1) * DIM + c4);
      *(float4*)(&Bld[r][c4]) = v;
    }
    __syncthreads();

    v8f c = {};
#pragma unroll
    for (int k0 = 0; k0 < DIM; k0 += 4) {
      v2f a, b;
      // A 16x4 layout: lane m holds row M=m; VGPR0/1 = K pair (k0+2*hi, k0+2*hi+1)
      a.x = Ald[m][k0 + 2 * hi];
      a.y = Ald[m][k0 + 2 * hi + 1];
      // B 4x16 layout mirrored: lane m holds col N=m (B[k][n] = bm[j0+n][k])
      b.x = Bld[m][k0 + 2 * hi];
      b.y = Bld[m][k0 + 2 * hi + 1];
      c = __builtin_amdgcn_wmma_f32_16x16x4_f32(false, a, false, b,
                                                (short)0, c, false, false);
    }

    // C layout: lane L col N=L%16, VGPR v row M = v + 8*(L/16)
#pragma unroll
    for (int v = 0; v < 8; ++v) sc[v + 8 * hi][m] = c[v];
    __syncthreads();

    if (lane < 16) {
#pragma unroll
      for (int n = 0; n < 16; ++n) {
        const float s = sc[lane][n];
        if (s > bestVal) { bestVal = s; bestIdx = j0 + n; }  // first-max tie break
      }
    }
    __syncthreads();
  }

  if (lane < 16) node_idx[(size_t)bb * pHalf + i0 + lane] = bestIdx;
}

// --------------------------------------------------------------------------
// Merge (r == pHalf for these shapes: every even token merges, unm is empty):
//   out[j] = (x[2j+1]*s[2j+1] + sum_{i: node_idx[i]==j} x[2i]*s[2i]) / s_new
// Deterministic gather (ascending source order), no float atomics.
// --------------------------------------------------------------------------
#define MAXHALF 2048
__global__ void tome_merge(const float* __restrict__ x,
                           const float* __restrict__ size_cur,  // null => all ones
                           const int*   __restrict__ node_idx,
                           float* __restrict__ x_next,
                           float* __restrict__ size_next,       // null => skip
                           int P) {
  const int pHalf = P >> 1;
  const int bb  = blockIdx.y;
  const int j   = blockIdx.x;        // destination (odd-set) index
  const int tid = threadIdx.x;

  __shared__ int   nd[MAXHALF];
  __shared__ int   list[MAXHALF];
  __shared__ float szs[MAXHALF];
  __shared__ float s_ssum;
  __shared__ int   s_cnt;

  for (int i = tid; i < pHalf; i += blockDim.x) {
    nd[i]  = node_idx[(size_t)bb * pHalf + i];
    szs[i] = size_cur ? size_cur[(size_t)bb * P + 2 * i] : 1.0f;
  }
  __syncthreads();

  if (tid == 0) {
    const float sd = size_cur ? size_cur[(size_t)bb * P + 2 * j + 1] : 1.0f;
    float ss = sd;
    int cnt = 0;
    for (int i = 0; i < pHalf; ++i)
      if (nd[i] == j) { list[cnt++] = i; ss += szs[i]; }
    s_cnt  = cnt;
    s_ssum = ss;
    if (size_next) size_next[(size_t)bb * pHalf + j] = ss;
  }
  __syncthreads();

  const float sd  = size_cur ? size_cur[(size_t)bb * P + 2 * j + 1] : 1.0f;
  const float inv = 1.0f / s_ssum;
  const int   cnt = s_cnt;
  const float* xb = x + (size_t)bb * P * CCH;
  float*       ob = x_next + ((size_t)bb * pHalf + j) * CCH;

  for (int c0 = tid * 4; c0 < CCH; c0 += blockDim.x * 4) {
    float4 acc = *(const float4*)(xb + (size_t)(2 * j + 1) * CCH + c0);
    acc.x *= sd; acc.y *= sd; acc.z *= sd; acc.w *= sd;
    for (int k = 0; k < cnt; ++k) {
      const int   i  = list[k];
      const float si = szs[i];
      const float4 v = *(const float4*)(xb + (size_t)(2 * i) * CCH + c0);
      acc.x += v.x * si; acc.y += v.y * si; acc.z += v.z * si; acc.w += v.w * si;
    }
    float4 o;
    o.x = acc.x * inv; o.y = acc.y * inv; o.z = acc.z * inv; o.w = acc.w * inv;
    *(float4*)(ob + c0) = o;
  }
}

// --------------------------------------------------------------------------
extern "C" void kernel_launch(void* const* d_in, const int* in_sizes, int n_in,
                              void* d_out, int out_size, void* d_ws, size_t ws_size,
                              hipStream_t stream) {
  (void)in_sizes; (void)n_in; (void)out_size; (void)ws_size;
  const float* x0 = (const float*)d_in[0];   // [8, 4096, 4096] fp32

  char* ws = (char*)d_ws;
  float* metric = (float*)ws;  ws += (size_t)NB * 4096 * DIM * sizeof(float);  // 16 MB
  int*   nidx   = (int*)ws;    ws += (size_t)NB * 2048 * sizeof(int);
  float* size1  = (float*)ws;  ws += (size_t)NB * 2048 * sizeof(float);
  float* x1     = (float*)ws;                                                  // 256 MB

  // ---- step 1: P=4096, r=2048 (all even tokens merge; sizes are 1) ----
  tome_metric<<<NB * 4096, DIM, 0, stream>>>(x0, metric);
  tome_scores_argmax<<<dim3(2048 / 16, NB), 32, 0, stream>>>(metric, nidx, 4096);
  tome_merge<<<dim3(2048, NB), 256, 0, stream>>>(x0, nullptr, nidx, x1, size1, 4096);

  // ---- step 2: P=2048, r=1024 -> write final [8,1024,4096] into d_out ----
  tome_metric<<<NB * 2048, DIM, 0, stream>>>(x1, metric);
  tome_scores_argmax<<<dim3(1024 / 16, NB), 32, 0, stream>>>(metric, nidx, 2048);
  tome_merge<<<dim3(1024, NB), 256, 0, stream>>>(x1, size1, nidx, (float*)d_out,
                                                 nullptr, 2048);
}